// OuterProduct_90572270338244
// MI455X (gfx1250) — compile-verified
//
#include <hip/hip_runtime.h>

#define DIM 256
#define H 32
#define OUT 128
#define LSEQ 384
#define EPSV 1e-5f

typedef __attribute__((ext_vector_type(2))) float v2f;
typedef __attribute__((ext_vector_type(8))) float v8f;

// ---------------- Kernel 1: LayerNorm + A/B projections ----------------
// One block per sequence position i. 256 threads load the row, reduce
// mean/var in LDS, normalize, then 64 threads compute the 32+32 dots.
__global__ __launch_bounds__(256) void ln_proj_kernel(
    const float* __restrict__ x, const float* __restrict__ gamma,
    const float* __restrict__ beta, const float* __restrict__ Wa,
    const float* __restrict__ Wb, float* __restrict__ a_out,
    float* __restrict__ b_out) {
  __shared__ float red[DIM];
  __shared__ float xn[DIM];
  const int i = blockIdx.x, t = threadIdx.x;
  const float v = x[i * DIM + t];
  red[t] = v;
  __syncthreads();
  for (int s = DIM / 2; s > 0; s >>= 1) {
    if (t < s) red[t] += red[t + s];
    __syncthreads();
  }
  const float mu = red[0] * (1.0f / DIM);
  __syncthreads();
  const float d = v - mu;
  red[t] = d * d;
  __syncthreads();
  for (int s = DIM / 2; s > 0; s >>= 1) {
    if (t < s) red[t] += red[t + s];
    __syncthreads();
  }
  const float rstd = rsqrtf(red[0] * (1.0f / DIM) + EPSV);
  xn[t] = d * rstd * gamma[t] + beta[t];
  __syncthreads();
  if (t < 2 * H) {
    const int h = t & (H - 1);
    const float* W = (t < H) ? Wa : Wb;
    float s = 0.f;
#pragma unroll 8
    for (int c = 0; c < DIM; ++c) s += xn[c] * W[h * DIM + c];
    float* o = (t < H) ? a_out : b_out;
    o[i * H + h] = s;
  }
}

// ---------------- Kernel 2a: transpose Wo -> W2[(c*32+d)*128 + o] ----------
__global__ __launch_bounds__(256) void transpose_wo(
    const float* __restrict__ Wo, float* __restrict__ W2) {
  const int idx = blockIdx.x * 256 + threadIdx.x;  // 0 .. 128*1024-1
  const int o = idx >> 10;                         // /1024
  const int k = idx & 1023;                        // c*32+d
  W2[k * OUT + o] = Wo[idx];
}

// ---------------- Kernel 2b: t2[i][d][o] = sum_c a[i,c]*Wo[o,c*32+d] -------
// W2 is laid out so consecutive threads read consecutive addresses.
__global__ __launch_bounds__(256) void make_t(
    const float* __restrict__ a_in, const float* __restrict__ W2,
    float* __restrict__ t2) {
  __shared__ float as[H];
  const int i = blockIdx.x, t = threadIdx.x;
  if (t < H) as[t] = a_in[i * H + t];
  __syncthreads();
#pragma unroll
  for (int r = 0; r < 16; ++r) {
    const int idx = t + 256 * r;  // = d*128 + o
    float s = 0.f;
#pragma unroll
    for (int c = 0; c < H; ++c) s += as[c] * W2[c * (H * OUT) + idx];
    t2[(size_t)i * (H * OUT) + idx] = s;
  }
}

// ---------------- Kernel 3: z[i,j,o] = sum_d b[j,d]*t[i,o,d] + bo[o] -------
// One block per i; 8 waves, wave w owns o-tile w (16 cols), loops 24 j-tiles.
// Each 16x16 tile accumulates K=32 via 8x V_WMMA_F32_16X16X4_F32.
// A frag (16x4 f32): lane<16 row=lane, V0=K0,V1=K1; lane>=16 row=lane-16,
// V0=K2,V1=K3.  B frag (4x16 f32): mirrored. C/D: V r -> row r + 8*(lane>=16),
// col = lane&15.
__global__ __launch_bounds__(256) void outer_wmma(
    const float* __restrict__ b_in, const float* __restrict__ t2,
    const float* __restrict__ bo, float* __restrict__ z) {
  __shared__ float bl[LSEQ * H];   // 48 KB: b[j][d]
  __shared__ float tl[H * OUT];    // 16 KB: t_i[d][o]
  const int i = blockIdx.x, tid = threadIdx.x;
  for (int idx = tid; idx < LSEQ * H; idx += 256) bl[idx] = b_in[idx];
  const float* ti = t2 + (size_t)i * (H * OUT);
  for (int idx = tid; idx < H * OUT; idx += 256) tl[idx] = ti[idx];
  __syncthreads();

  const int lane = tid & 31;
  const int wv = tid >> 5;       // o-tile index 0..7
  const int half = lane >> 4;    // 0 or 1 (K/M split per ISA layout)
  const int lr = lane & 15;
  const int o0 = wv * 16;
  const float bias = bo[o0 + lr];
  float* zbase = z + (size_t)i * LSEQ * OUT + o0 + lr;

  for (int jt = 0; jt < LSEQ / 16; ++jt) {
    const int j0 = jt * 16;
    v8f acc = {};
#pragma unroll
    for (int kk = 0; kk < H / 4; ++kk) {
      const int kb = kk * 4 + half * 2;
      v2f af, bf;
      const float* ap = &bl[(j0 + lr) * H + kb];
      af.x = ap[0];
      af.y = ap[1];
      bf.x = tl[kb * OUT + o0 + lr];
      bf.y = tl[(kb + 1) * OUT + o0 + lr];
      acc = __builtin_amdgcn_wmma_f32_16x16x4_f32(
          /*neg_a=*/false, af, /*neg_b=*/false, bf,
          /*c_mod=*/(short)0, acc, /*reuse_a=*/false, /*reuse_b=*/false);
    }
    float* zr = zbase + (size_t)j0 * OUT;
#pragma unroll
    for (int r = 0; r < 8; ++r) {
      zr[(size_t)(r + half * 8) * OUT] = acc[r] + bias;
    }
  }
}

extern "C" void kernel_launch(void* const* d_in, const int* in_sizes, int n_in,
                              void* d_out, int out_size, void* d_ws,
                              size_t ws_size, hipStream_t stream) {
  const float* x     = (const float*)d_in[0];
  const float* gamma = (const float*)d_in[1];
  const float* beta  = (const float*)d_in[2];
  const float* Wa    = (const float*)d_in[3];
  const float* Wb    = (const float*)d_in[4];
  const float* Wo    = (const float*)d_in[5];
  const float* bo    = (const float*)d_in[6];
  float* z = (float*)d_out;

  float* a_buf = (float*)d_ws;              // 384*32
  float* b_buf = a_buf + LSEQ * H;          // 384*32
  float* W2    = b_buf + LSEQ * H;          // 1024*128
  float* t2    = W2 + (H * H) * OUT;        // 384*32*128

  ln_proj_kernel<<<LSEQ, 256, 0, stream>>>(x, gamma, beta, Wa, Wb, a_buf, b_buf);
  transpose_wo<<<(OUT * H * H) / 256, 256, 0, stream>>>(Wo, W2);
  make_t<<<LSEQ, 256, 0, stream>>>(a_buf, W2, t2);
  outer_wmma<<<LSEQ, 256, 0, stream>>>(b_buf, t2, bo, z);
}